// GraphLevelEncoder_7232724927021
// MI455X (gfx1250) — compile-verified
//
#include <hip/hip_runtime.h>
#include <hip/hip_bf16.h>

typedef float v2f __attribute__((ext_vector_type(2)));
typedef float v8f __attribute__((ext_vector_type(8)));

#define H64 64
#define BN_EPS 1e-5f

// ---------------- degree / norm ----------------
__global__ void deg_init(float* __restrict__ deg, int N) {
  int i = blockIdx.x * blockDim.x + threadIdx.x;
  if (i < N) deg[i] = 1.0f;  // self-loop weight
}

__global__ void deg_scatter(const long long* __restrict__ dst,
                            const float* __restrict__ ew,
                            float* __restrict__ deg, long long E) {
  long long e = (long long)blockIdx.x * blockDim.x + threadIdx.x;
  if (e < E) atomicAdd(&deg[(int)dst[e]], ew[e]);
}

__global__ void deg_rsqrt(float* __restrict__ deg, int N) {
  int i = blockIdx.x * blockDim.x + threadIdx.x;
  if (i < N) deg[i] = rsqrtf(deg[i]);
}

// ---------------- GEMM: H = X[N,64] @ W[64,64] via V_WMMA_F32_16X16X4_F32 ----
// One wave per 16-row strip; 4 column tiles of 16; K-loop of 16 steps of 4.
__global__ void gemm64_wmma(const float* __restrict__ X,
                            const float* __restrict__ W,
                            float* __restrict__ Hout, int nTiles) {
  int wave = threadIdx.x >> 5;
  int lane = threadIdx.x & 31;
  int tile = blockIdx.x * (blockDim.x >> 5) + wave;
  if (tile >= nTiles) return;

  int m = lane & 15;              // row within tile (A), col within tile (B/C/D)
  int khalf = (lane >> 4) << 1;   // 0 for lanes 0-15, 2 for lanes 16-31
  const float* xr = X + ((size_t)tile * 16 + m) * H64;

  v8f acc0 = {}, acc1 = {}, acc2 = {}, acc3 = {};
  #pragma unroll
  for (int k = 0; k < 64; k += 4) {
    // A 16x4 fp32 layout: VGPR0 = K=khalf, VGPR1 = K=khalf+1
    v2f a;
    a.x = xr[k + khalf];
    a.y = xr[k + khalf + 1];
    // B 4x16 fp32 layout: VGPR0 = row K=khalf (N striped on lanes), VGPR1 = K=khalf+1
    const float* wr0 = W + (size_t)(k + khalf) * H64 + m;
    const float* wr1 = wr0 + H64;
    v2f b0; b0.x = wr0[0];  b0.y = wr1[0];
    v2f b1; b1.x = wr0[16]; b1.y = wr1[16];
    v2f b2; b2.x = wr0[32]; b2.y = wr1[32];
    v2f b3; b3.x = wr0[48]; b3.y = wr1[48];
    acc0 = __builtin_amdgcn_wmma_f32_16x16x4_f32(false, a, false, b0, (short)0, acc0, false, false);
    acc1 = __builtin_amdgcn_wmma_f32_16x16x4_f32(false, a, false, b1, (short)0, acc1, false, false);
    acc2 = __builtin_amdgcn_wmma_f32_16x16x4_f32(false, a, false, b2, (short)0, acc2, false, false);
    acc3 = __builtin_amdgcn_wmma_f32_16x16x4_f32(false, a, false, b3, (short)0, acc3, false, false);
  }
  // C/D 16x16 fp32: VGPR v, lane l -> M = v + 8*(l>=16), Ncol = l&15
  float* hp = Hout + ((size_t)tile * 16 + ((lane >> 4) << 3)) * H64 + m;
  #pragma unroll
  for (int v = 0; v < 8; ++v) {
    hp[v * H64 + 0]  = acc0[v];
    hp[v * H64 + 16] = acc1[v];
    hp[v * H64 + 32] = acc2[v];
    hp[v * H64 + 48] = acc3[v];
  }
}

// Scalar tail for N % 16 rows (not taken for N=100000).
__global__ void gemm64_tail(const float* __restrict__ X, const float* __restrict__ W,
                            float* __restrict__ Hout, int rowStart, int N) {
  int idx = blockIdx.x * blockDim.x + threadIdx.x;
  int row = rowStart + (idx >> 6);
  int col = idx & 63;
  if (row >= N) return;
  float acc = 0.f;
  for (int k = 0; k < H64; ++k) acc = fmaf(X[(size_t)row * H64 + k], W[(size_t)k * H64 + col], acc);
  Hout[(size_t)row * H64 + col] = acc;
}

// ---------------- aggregation ----------------
// Initialize accumulator with self-loop term: out[i] = h[i] * dinv[i]^2
__global__ void selfloop_init(const float* __restrict__ h, const float* __restrict__ dinv,
                              float* __restrict__ out, int N) {
  int i = blockIdx.x * blockDim.x + threadIdx.x;     // N*16 float4 slots
  if (i >= N * 16) return;
  int node = i >> 4, c = i & 15;
  float di = dinv[node];
  float nrm = di * di;
  float4 v = ((const float4*)(h + (size_t)node * H64))[c];
  float4 o = make_float4(v.x * nrm, v.y * nrm, v.z * nrm, v.w * nrm);
  ((float4*)(out + (size_t)node * H64))[c] = o;
}

// 16 lanes (half wave32) per edge; lane 0 computes norm, broadcast via shfl.
__global__ void edge_scatter(const long long* __restrict__ src,
                             const long long* __restrict__ dst,
                             const float* __restrict__ ew,
                             const float* __restrict__ dinv,
                             const float* __restrict__ h,
                             float* __restrict__ out, long long E) {
  long long e = (long long)blockIdx.x * (blockDim.x >> 4) + (threadIdx.x >> 4);
  int lane16 = threadIdx.x & 15;
  if (e >= E) return;
  int s = 0, d = 0;
  float nrm = 0.f;
  if (lane16 == 0) {
    s = (int)src[e];
    d = (int)dst[e];
    nrm = dinv[s] * ew[e] * dinv[d];
  }
  s   = __shfl(s,   0, 16);
  d   = __shfl(d,   0, 16);
  nrm = __shfl(nrm, 0, 16);
  float4 v = ((const float4*)(h + (size_t)s * H64))[lane16];
  float* o = out + (size_t)d * H64 + lane16 * 4;
  atomicAdd(o + 0, v.x * nrm);
  atomicAdd(o + 1, v.y * nrm);
  atomicAdd(o + 2, v.z * nrm);
  atomicAdd(o + 3, v.w * nrm);
}

// ---------------- batchnorm + relu ----------------
__global__ void stats_zero(float* __restrict__ stats) {
  if (threadIdx.x < 128) stats[threadIdx.x] = 0.f;
}

__global__ void bn_stats(const float* __restrict__ h, float* __restrict__ stats, int N) {
  __shared__ float ssum[64];
  __shared__ float ssq[64];
  if (threadIdx.x < 64) { ssum[threadIdx.x] = 0.f; ssq[threadIdx.x] = 0.f; }
  __syncthreads();
  float s = 0.f, q = 0.f;
  size_t total = (size_t)N * H64;
  size_t stride = (size_t)gridDim.x * blockDim.x;       // multiple of 64
  for (size_t i = (size_t)blockIdx.x * blockDim.x + threadIdx.x; i < total; i += stride) {
    float v = h[i];
    s += v;
    q += v * v;
  }
  int f = threadIdx.x & 63;                             // fixed per thread
  atomicAdd(&ssum[f], s);
  atomicAdd(&ssq[f], q);
  __syncthreads();
  if (threadIdx.x < 64) {
    atomicAdd(&stats[threadIdx.x], ssum[threadIdx.x]);
    atomicAdd(&stats[64 + threadIdx.x], ssq[threadIdx.x]);
  }
}

__global__ void bn_relu(const float* __restrict__ h, const float* __restrict__ stats,
                        const float* __restrict__ gamma, const float* __restrict__ beta,
                        float* __restrict__ out, int N) {
  size_t i = (size_t)blockIdx.x * blockDim.x + threadIdx.x;
  if (i >= (size_t)N * H64) return;
  int f = (int)(i & 63);
  float invN = 1.0f / (float)N;
  float mean = stats[f] * invN;
  float var = stats[64 + f] * invN - mean * mean;
  float sc = rsqrtf(var + BN_EPS) * gamma[f];
  float v = (h[i] - mean) * sc + beta[f];
  out[i] = v > 0.f ? v : 0.f;
}

// ---------------- host side ----------------
extern "C" void kernel_launch(void* const* d_in, const int* in_sizes, int n_in,
                              void* d_out, int out_size, void* d_ws, size_t ws_size,
                              hipStream_t stream) {
  const float*     x    = (const float*)d_in[0];
  const long long* ei   = (const long long*)d_in[1];
  const float*     ew   = (const float*)d_in[2];
  const float*     W1   = (const float*)d_in[3];
  const float*     g1   = (const float*)d_in[5];
  const float*     be1  = (const float*)d_in[6];
  const float*     W2   = (const float*)d_in[7];
  const float*     g2   = (const float*)d_in[9];
  const float*     be2  = (const float*)d_in[10];
  (void)n_in; (void)out_size; (void)ws_size;

  int N = in_sizes[0] / H64;
  long long E = in_sizes[2];
  const long long* srcp = ei;
  const long long* dstp = ei + E;

  char* ws = (char*)d_ws;
  size_t o = 0;
  float* dinv = (float*)(ws + o); o += (size_t)N * 4;           o = (o + 255) & ~(size_t)255;
  float* bufA = (float*)(ws + o); o += (size_t)N * H64 * 4;     o = (o + 255) & ~(size_t)255;
  float* bufB = (float*)(ws + o); o += (size_t)N * H64 * 4;     o = (o + 255) & ~(size_t)255;
  float* stats = (float*)(ws + o);

  const int T = 256;
  int gN      = (N + T - 1) / T;
  int gE      = (int)((E + T - 1) / T);
  int gEdge   = (int)((E + 15) / 16);                 // 16 edges per 256-thread block
  int gNode16 = (N * 16 + T - 1) / T;
  int gElem   = (int)(((size_t)N * H64 + T - 1) / T);
  int nTiles  = N / 16;
  int gGemm   = (nTiles + 7) / 8;                     // 8 waves per block
  int rem     = N - nTiles * 16;
  int gStats  = 1024;

  // --- degree / symmetric norm ---
  deg_init<<<gN, T, 0, stream>>>(dinv, N);
  deg_scatter<<<gE, T, 0, stream>>>(dstp, ew, dinv, E);
  deg_rsqrt<<<gN, T, 0, stream>>>(dinv, N);

  // --- layer 1 ---
  gemm64_wmma<<<gGemm, T, 0, stream>>>(x, W1, bufA, nTiles);
  if (rem) gemm64_tail<<<(rem * 64 + T - 1) / T, T, 0, stream>>>(x, W1, bufA, nTiles * 16, N);
  selfloop_init<<<gNode16, T, 0, stream>>>(bufA, dinv, bufB, N);
  edge_scatter<<<gEdge, T, 0, stream>>>(srcp, dstp, ew, dinv, bufA, bufB, E);
  stats_zero<<<1, 128, 0, stream>>>(stats);
  bn_stats<<<gStats, T, 0, stream>>>(bufB, stats, N);
  bn_relu<<<gElem, T, 0, stream>>>(bufB, stats, g1, be1, bufA, N);

  // --- layer 2 ---
  gemm64_wmma<<<gGemm, T, 0, stream>>>(bufA, W2, bufB, nTiles);
  if (rem) gemm64_tail<<<(rem * 64 + T - 1) / T, T, 0, stream>>>(bufA, W2, bufB, nTiles * 16, N);
  selfloop_init<<<gNode16, T, 0, stream>>>(bufB, dinv, bufA, N);
  edge_scatter<<<gEdge, T, 0, stream>>>(srcp, dstp, ew, dinv, bufB, bufA, E);
  stats_zero<<<1, 128, 0, stream>>>(stats);
  bn_stats<<<gStats, T, 0, stream>>>(bufA, stats, N);
  bn_relu<<<gElem, T, 0, stream>>>(bufA, stats, g2, be2, (float*)d_out, N);
}